// ECELoss_20263655702825
// MI455X (gfx1250) — compile-verified
//
#include <hip/hip_runtime.h>
#include <math.h>

#define C 256
#define NBINS 15
#define ROWS_PER_TILE 32
#define TILES_PER_BLOCK 16
#define ROWS_PER_BLOCK (ROWS_PER_TILE * TILES_PER_BLOCK) // 512
#define NBUF 3                                           // triple buffer, depth-2 prefetch
#define THREADS 256                                      // 8 waves (wave32)

typedef unsigned int u32x4 __attribute__((ext_vector_type(4)));
typedef int i32x8 __attribute__((ext_vector_type(8)));
typedef int i32x4 __attribute__((ext_vector_type(4)));

// Issue a TDM load of a [rows x 256] fp32 tile (row-major, stride 256) from
// global address `gaddr` into LDS byte-offset `lds_off`.
// D# layout per CDNA5 ISA ch.8 (group0: count/lds_addr/global_addr/type,
// group1: data_size/tensor dims/tile dims/strides; groups 2-3 zero for 2D).
__device__ __forceinline__ void tdm_load_tile(unsigned lds_off,
                                              unsigned long long gaddr,
                                              int rows) {
  u32x4 g0;
  g0[0] = 1u;                                      // count=1 (valid), user mode
  g0[1] = lds_off;                                 // lds_addr (bytes)
  g0[2] = (unsigned)(gaddr & 0xFFFFFFFFu);         // global_addr[31:0]
  g0[3] = (unsigned)((gaddr >> 32) & 0x1FFFFFFu)   // global_addr[56:32]
          | (2u << 30);                            // type=2 ("image")
  i32x8 g1;
  g1[0] = (int)(2u << 16);                         // data_size=2 -> 4 bytes
  g1[1] = (int)((unsigned)C << 16);                // tensor_dim0[15:0]=256
  g1[2] = (int)((unsigned)rows << 16);             // tensor_dim1[15:0]=rows
  g1[3] = (int)((unsigned)C << 16);                // tile_dim0=256
  g1[4] = rows;                                    // tile_dim1=rows, tile_dim2=0
  g1[5] = C;                                       // tensor_dim0_stride=256
  g1[6] = 0;
  g1[7] = 0;
  i32x4 g2 = {0, 0, 0, 0};
  i32x4 g3 = {0, 0, 0, 0};
#if defined(__clang_major__) && (__clang_major__ >= 23)
  i32x8 g4 = {0, 0, 0, 0, 0, 0, 0, 0};
  __builtin_amdgcn_tensor_load_to_lds(g0, g1, g2, g3, g4, 0);
#else
  __builtin_amdgcn_tensor_load_to_lds(g0, g1, g2, g3, 0);
#endif
}

__global__ void ece_zero_kernel(double* __restrict__ g) {
  int i = threadIdx.x;
  if (i < 3 * NBINS) g[i] = 0.0;
}

__global__ __launch_bounds__(THREADS) void ece_main_kernel(
    const float* __restrict__ probs, const long long* __restrict__ labels,
    int n_rows, double* __restrict__ gbins) {
  __shared__ __align__(16) float buf[NBUF][ROWS_PER_TILE * C]; // 96 KB
  __shared__ float s_bins[3 * NBINS]; // [0..14]=num [15..29]=conf [30..44]=acc

  const int tid = threadIdx.x;
  const int wave = tid >> 5;
  const int lane = tid & 31;

  if (tid < 3 * NBINS) s_bins[tid] = 0.0f;

  const long long row0 = (long long)blockIdx.x * ROWS_PER_BLOCK;
  long long rows_left = (long long)n_rows - row0;
  if (rows_left <= 0) return; // uniform; grid sized so every block has rows
  const int rows_block =
      rows_left < ROWS_PER_BLOCK ? (int)rows_left : ROWS_PER_BLOCK;
  const int T = (rows_block + ROWS_PER_TILE - 1) / ROWS_PER_TILE;

  // LDS byte offsets (flat-addr low 32 bits == LDS offset per aperture map).
  unsigned lds_buf[NBUF];
#pragma unroll
  for (int i = 0; i < NBUF; ++i) lds_buf[i] = (unsigned)(size_t)(&buf[i][0]);

  const unsigned long long gbase =
      (unsigned long long)(size_t)probs + (unsigned long long)row0 * C * 4ull;

  const bool issuer = (wave == 0);

  auto tile_rows = [&](int t) {
    int r = rows_block - t * ROWS_PER_TILE;
    return r < ROWS_PER_TILE ? r : ROWS_PER_TILE;
  };
  auto issue = [&](int t) {
    tdm_load_tile(lds_buf[t % NBUF],
                  gbase + (unsigned long long)t * (ROWS_PER_TILE * C * 4ull),
                  tile_rows(t));
  };

  // Prime: two tiles in flight.
  if (issuer) {
    issue(0);
    if (T > 1) issue(1);
  }

  for (int t = 0; t < T; ++t) {
    // At loop top, tiles issued are 0..min(t+1, T-1); completion is in-order,
    // so leaving min(1, T-1-t) outstanding guarantees tile t has landed.
    if (issuer) {
      if (t + 1 < T) __builtin_amdgcn_s_wait_tensorcnt(1);
      else           __builtin_amdgcn_s_wait_tensorcnt(0);
    }
    // Publishes tile t to all waves AND proves all waves finished reading
    // tile t-1 (whose buffer tile t+2 will overwrite). Also covers s_bins
    // zero-init on the first iteration.
    __syncthreads();
    if (issuer && (t + 2 < T)) issue(t + 2);

    const float* tile = &buf[t % NBUF][0];
    const int rt = tile_rows(t);

    for (int r = wave; r < rt; r += (THREADS / 32)) {
      const float4* rowp = (const float4*)(tile + r * C);
      float4 a = rowp[lane];
      float4 b = rowp[lane + 32];
      float m = fmaxf(fmaxf(fmaxf(a.x, a.y), fmaxf(a.z, a.w)),
                      fmaxf(fmaxf(b.x, b.y), fmaxf(b.z, b.w)));
#pragma unroll
      for (int off = 16; off > 0; off >>= 1)
        m = fmaxf(m, __shfl_xor(m, off, 32));
      if (lane == 0) {
        long long grow = row0 + (long long)t * ROWS_PER_TILE + r;
        int lab = (int)labels[grow];
        float p = tile[r * C + lab];           // prob at label, from LDS tile
        float acc = (p == m) ? 1.0f : 0.0f;    // pred==label (ties ~0 prob)
        int bidx = (int)ceilf(m * (float)NBINS) - 1;
        bidx = bidx < 0 ? 0 : (bidx > NBINS - 1 ? NBINS - 1 : bidx);
        atomicAdd(&s_bins[bidx], 1.0f);            // count
        atomicAdd(&s_bins[NBINS + bidx], m);       // conf sum
        atomicAdd(&s_bins[2 * NBINS + bidx], acc); // acc sum
      }
    }
  }

  __syncthreads(); // all waves done accumulating into s_bins
  if (tid < 3 * NBINS) {
    float v = s_bins[tid];
    if (v != 0.0f) atomicAdd(&gbins[tid], (double)v);
  }
}

__global__ void ece_final_kernel(const double* __restrict__ gbins,
                                 float* __restrict__ out) {
  __shared__ double gaps[NBINS];
  int b = threadIdx.x;
  if (b < NBINS) {
    double num = gbins[b];
    double conf = gbins[NBINS + b];
    double acc = gbins[2 * NBINS + b];
    gaps[b] = (num > 0.0) ? fabs(conf - acc) : 0.0;
    out[1 + b] = (float)acc;          // acc_sum (correct_list)
    out[1 + NBINS + b] = (float)num;  // num_list
  }
  __syncthreads();
  if (b == 0) {
    double s = 0.0;
#pragma unroll
    for (int i = 0; i < NBINS; ++i) s += gaps[i];
    out[0] = (float)s; // ece
  }
}

extern "C" void kernel_launch(void* const* d_in, const int* in_sizes, int n_in,
                              void* d_out, int out_size, void* d_ws,
                              size_t ws_size, hipStream_t stream) {
  const float* probs = (const float*)d_in[0];
  const long long* labels = (const long long*)d_in[1]; // int64 labels
  (void)n_in; (void)out_size; (void)ws_size;
  const int n_rows = in_sizes[1]; // N = label count; probs is N x 256

  double* gbins = (double*)d_ws; // 45 fp64 accumulators

  ece_zero_kernel<<<1, 64, 0, stream>>>(gbins);
  const int grid = (n_rows + ROWS_PER_BLOCK - 1) / ROWS_PER_BLOCK;
  ece_main_kernel<<<grid, THREADS, 0, stream>>>(probs, labels, n_rows, gbins);
  ece_final_kernel<<<1, 32, 0, stream>>>(gbins, (float*)d_out);
}